// DetectionLayer_25580825215449
// MI455X (gfx1250) — compile-verified
//
#include <hip/hip_runtime.h>
#include <float.h>
#include <math.h>

// Problem constants (match reference)
#define B_      16
#define N_      2000
#define C_      81
#define MAXI    100
#define MIN_CONF 0.7f
#define NMS_T    0.3f
#define CLS_OFF  2.0f

#define M_  2048   // padded sort size (power of two >= N_)
#define BT  1024   // threads per sort/NMS workgroup (32 waves on gfx1250)

#ifndef __has_builtin
#define __has_builtin(x) 0
#endif

// CDNA5 async global->LDS path (per-lane gather into LDS, ASYNCcnt-tracked).
// Guarded so the host pass / older toolchains fall back to plain ds stores.
#if defined(__HIP_DEVICE_COMPILE__) && defined(__gfx1250__) && \
    __has_builtin(__builtin_amdgcn_global_load_async_to_lds_b128) && \
    __has_builtin(__builtin_amdgcn_s_wait_asynccnt)
#define USE_ASYNC_GATHER 1
#endif

#ifdef USE_ASYNC_GATHER
// The builtin's parameter type (from clang diagnostic):
//   int __attribute__((vector_size(16))) addrspace(1)* / addrspace(3)*
typedef int v4i_async __attribute__((vector_size(16)));
typedef __attribute__((address_space(1))) v4i_async* g_v4i_ptr;
typedef __attribute__((address_space(3))) v4i_async* l_v4i_ptr;
#endif

// ---------------------------------------------------------------------------
// Kernel 1: per-ROI class argmax + class-specific delta refinement + clip.
// Bandwidth-bound: ~11.5 MB touched -> ~0.5us at 23.3 TB/s.
// ---------------------------------------------------------------------------
__global__ void refine_kernel(const float* __restrict__ rois,
                              const float* __restrict__ probs,
                              const float* __restrict__ deltas,
                              float* __restrict__ ref,      // [B*N,4] refined boxes
                              float* __restrict__ cls_out,  // [B*N] class id (float)
                              float* __restrict__ key_out)  // [B*N] nms score key
{
    const int idx = blockIdx.x * blockDim.x + threadIdx.x;
    if (idx >= B_ * N_) return;

    // argmax over 81 classes (first-max wins, matching jnp.argmax)
    const float* p = probs + (size_t)idx * C_;
    float best = p[0];
    int   cls  = 0;
    for (int c = 1; c < C_; ++c) {
        const float v = p[c];
        if (v > best) { best = v; cls = c; }
    }

    // class-specific deltas: each class record is exactly 16B -> aligned B128 load
    const float4 d4 = *(const float4*)(deltas + ((size_t)idx * C_ + cls) * 4);
    const float dy = d4.x * 0.1f;
    const float dx = d4.y * 0.1f;
    const float dh = d4.z * 0.2f;
    const float dw = d4.w * 0.2f;

    const float4 r4 = *(const float4*)(rois + (size_t)idx * 4);  // y1,x1,y2,x2
    float h  = r4.z - r4.x;
    float w  = r4.w - r4.y;
    float cy = r4.x + 0.5f * h + dy * h;
    float cx = r4.y + 0.5f * w + dx * w;
    h *= expf(dh);
    w *= expf(dw);

    const float y1 = fminf(fmaxf(cy - 0.5f * h, 0.0f), 1.0f);
    const float x1 = fminf(fmaxf(cx - 0.5f * w, 0.0f), 1.0f);
    const float y2 = fminf(fmaxf(cy + 0.5f * h, 0.0f), 1.0f);
    const float x2 = fminf(fmaxf(cx + 0.5f * w, 0.0f), 1.0f);

    *(float4*)(ref + (size_t)idx * 4) = make_float4(y1, x1, y2, x2);
    cls_out[idx] = (float)cls;

    const bool valid = (cls > 0) && (best >= MIN_CONF);
    key_out[idx] = valid ? best : -1.0f;
}

// ---------------------------------------------------------------------------
// Kernel 2: per-batch bitonic sort (desc score, asc index tiebreak == stable
// argsort(-score)), greedy NMS on class-offset boxes, top-100 emit.
// One 1024-thread workgroup per batch; everything in LDS (56KB of 320KB/WGP).
// ---------------------------------------------------------------------------
__global__ __launch_bounds__(BT) void sort_nms_kernel(
        const float* __restrict__ ref,
        const float* __restrict__ cls,
        const float* __restrict__ key,
        float* __restrict__ out)
{
    __shared__ float  s_key[M_];
    __shared__ int    s_idx[M_];
    __shared__ float4 s_box[M_];   // class-offset boxes (y1,x1,y2,x2)
    __shared__ int    s_keep[M_];

    const int b   = blockIdx.x;
    const int tid = threadIdx.x;
    const float* bkey = key + (size_t)b * N_;

    // init: real entries + pad with -FLT_MAX so padding sinks to the tail
    for (int p = tid; p < M_; p += BT) {
        s_key[p]  = (p < N_) ? bkey[p] : -FLT_MAX;
        s_idx[p]  = p;
        s_keep[p] = 1;
    }
    __syncthreads();

    // bitonic sort: descending key, ascending original index on ties
    for (unsigned k = 2; k <= M_; k <<= 1) {
        for (unsigned j = k >> 1; j > 0; j >>= 1) {
            const unsigned t  = (unsigned)tid;
            const unsigned i0 = 2u * t - (t & (j - 1u));
            const unsigned i1 = i0 + j;
            const float ka = s_key[i0], kb = s_key[i1];
            const int   ia = s_idx[i0], ib = s_idx[i1];
            const bool aBefore = (ka > kb) || ((ka == kb) && (ia < ib));
            const bool up = ((i0 & k) == 0u);
            if (up ? !aBefore : aBefore) {
                s_key[i0] = kb; s_key[i1] = ka;
                s_idx[i0] = ib; s_idx[i1] = ia;
            }
            __syncthreads();
        }
    }

    // gather refined boxes into LDS in sorted order.
    // CDNA5 path: per-lane async global->LDS B128 gather (ASYNCcnt).
    for (int p = tid; p < M_; p += BT) {
        if (p < N_) {
            const int g = s_idx[p];
            const float* src = ref + ((size_t)b * N_ + g) * 4;
#ifdef USE_ASYNC_GATHER
            __builtin_amdgcn_global_load_async_to_lds_b128(
                (g_v4i_ptr)src,
                (l_v4i_ptr)&s_box[p],
                0, 0);
#else
            s_box[p] = *(const float4*)src;
#endif
        } else {
            s_box[p] = make_float4(0.0f, 0.0f, 0.0f, 0.0f);
        }
    }
#ifdef USE_ASYNC_GATHER
    __builtin_amdgcn_s_wait_asynccnt(0);
#endif
    __syncthreads();

    // add class offset (separates classes so cross-class IoU == 0)
    for (int p = tid; p < N_; p += BT) {
        const float co = cls[(size_t)b * N_ + s_idx[p]] * CLS_OFF;
        float4 v = s_box[p];
        v.x += co; v.y += co; v.z += co; v.w += co;
        s_box[p] = v;
    }
    __syncthreads();

    // greedy NMS: sequential over i, parallel over j. Barrier only when i is
    // still kept (the branch is uniform: same LDS word read by all lanes).
    for (int i = 0; i < N_ - 1; ++i) {
        if (s_keep[i]) {
            const float4 bi = s_box[i];
            const float  ai = (bi.z - bi.x) * (bi.w - bi.y);
            for (int jj = i + 1 + tid; jj < N_; jj += BT) {
                const float4 bj = s_box[jj];
                const float  aj = (bj.z - bj.x) * (bj.w - bj.y);
                float ih = fminf(bi.z, bj.z) - fmaxf(bi.x, bj.x);
                float iw = fminf(bi.w, bj.w) - fmaxf(bi.y, bj.y);
                ih = fmaxf(ih, 0.0f);
                iw = fmaxf(iw, 0.0f);
                const float inter = ih * iw;
                const float uni   = fmaxf(ai + aj - inter, 1e-8f);
                if (inter > NMS_T * uni) s_keep[jj] = 0;   // iou > 0.3
            }
            __syncthreads();
        }
    }
    __syncthreads();

    // emit: first <=100 kept (score>0) entries in sorted order; zero the rest.
    if (tid == 0) {
        float* o = out + (size_t)b * MAXI * 6;
        int cnt = 0;
        for (int p = 0; p < N_ && cnt < MAXI; ++p) {
            if (s_keep[p] && s_key[p] > 0.0f) {
                const int g = s_idx[p];
                const float4 rb = *(const float4*)(ref + ((size_t)b * N_ + g) * 4);
                float* r = o + cnt * 6;
                r[0] = rb.x; r[1] = rb.y; r[2] = rb.z; r[3] = rb.w;
                r[4] = cls[(size_t)b * N_ + g];
                r[5] = s_key[p];
                ++cnt;
            }
        }
        for (; cnt < MAXI; ++cnt) {
            float* r = o + cnt * 6;
            r[0] = r[1] = r[2] = r[3] = r[4] = r[5] = 0.0f;
        }
    }
}

// ---------------------------------------------------------------------------
extern "C" void kernel_launch(void* const* d_in, const int* in_sizes, int n_in,
                              void* d_out, int out_size, void* d_ws, size_t ws_size,
                              hipStream_t stream) {
    (void)in_sizes; (void)n_in; (void)out_size; (void)ws_size;

    const float* rois   = (const float*)d_in[0];  // [B,N,4]
    const float* probs  = (const float*)d_in[1];  // [B,N,C]
    const float* deltas = (const float*)d_in[2];  // [B,N,C,4]
    float* out = (float*)d_out;                   // [B,100,6]

    float* ws      = (float*)d_ws;
    float* ws_ref  = ws;                          // B*N*4 = 128000 floats
    float* ws_cls  = ws_ref + (size_t)B_ * N_ * 4;// B*N   =  32000 floats
    float* ws_key  = ws_cls + (size_t)B_ * N_;    // B*N   =  32000 floats

    const int total = B_ * N_;
    refine_kernel<<<(total + 255) / 256, 256, 0, stream>>>(
        rois, probs, deltas, ws_ref, ws_cls, ws_key);

    sort_nms_kernel<<<B_, BT, 0, stream>>>(ws_ref, ws_cls, ws_key, out);
}